// QLSTM_65481071398748
// MI455X (gfx1250) — compile-verified
//
#include <hip/hip_runtime.h>
#include <math.h>

typedef __attribute__((ext_vector_type(2))) float v2f;
typedef __attribute__((ext_vector_type(8))) float v8f;

#define TT 512
#define BB 256
#define DD 256
#define HH 128
#define NQ 128
#define DH (DD + HH) /* 384 */
#define XS_LD 260    /* 64-row x-tile row stride: 260 % 64 == 4 -> conflict-free b64 reads */
#define HP 132       /* 16-row state-tile row stride: 132 % 64 == 4 -> conflict-free */

__device__ __forceinline__ float sigmoidf_(float x) {
    return 1.0f / (1.0f + __expf(-x));
}

// ---------------------------------------------------------------------------
// Kernel 1: time-parallel x-projections for all three gates.
// Block = 64 rows of (t*B+b), 768 threads = 24 waves; wave = (gate, n-tile).
// Each wave holds 4 M-tile accumulators (4 x v8f) and reuses every B fragment
// across the 4 M-tiles -> 4x less weight traffic from L2, 4 WMMA per B load.
// ---------------------------------------------------------------------------
__global__ __launch_bounds__(768, 1)
void qlstm_xproj(const float* __restrict__ X,
                 const float* __restrict__ Wp,
                 const float* __restrict__ Wu,
                 const float* __restrict__ Wo,
                 const float* __restrict__ bu,
                 const float* __restrict__ bo,
                 float* __restrict__ XP,
                 float* __restrict__ XU,
                 float* __restrict__ XO)
{
    extern __shared__ float xs[]; // [64][XS_LD] = 66,560 B
    const int tid  = threadIdx.x;
    const int row0 = blockIdx.x * 64;

    for (int i = tid; i < 64 * DD; i += 768) {
        int r = i >> 8;          // i / 256
        int c = i & 255;         // i % 256
        xs[r * XS_LD + c] = X[(size_t)(row0 + r) * DD + c];
    }
    __syncthreads();

    const int wave  = tid >> 5;          // 0..23
    const int g     = wave >> 3;         // gate 0..2
    const int nt    = wave & 7;          // n-tile 0..7
    const int lane  = tid & 31;
    const int n     = nt * 16 + (lane & 15);
    const int koff  = (lane < 16) ? 0 : 2;
    const int m     = lane & 15;
    const int mbase = (lane < 16) ? 0 : 8;

    const float* W = (g == 0) ? Wp : (g == 1) ? Wu : Wo;
    float bias     = (g == 0) ? 0.0f : (g == 1) ? bu[n] : bo[n];

    v8f acc[4];
#pragma unroll
    for (int mt = 0; mt < 4; ++mt)
#pragma unroll
        for (int r = 0; r < 8; ++r) acc[mt][r] = bias;

    for (int k0 = 0; k0 < DD; k0 += 4) {
        // B: 4x16 tile of W^T -> B[k][n] = W[n][k]; loaded once, used 4x.
        v2f b = *(const v2f*)&W[(size_t)n * DH + k0 + koff];
#pragma unroll
        for (int mt = 0; mt < 4; ++mt) {
            v2f a = *(const v2f*)&xs[(mt * 16 + m) * XS_LD + k0 + koff];
            acc[mt] = __builtin_amdgcn_wmma_f32_16x16x4_f32(
                false, a, false, b, (short)0, acc[mt], false, false);
        }
    }

    float* O = (g == 0) ? XP : (g == 1) ? XU : XO;
#pragma unroll
    for (int mt = 0; mt < 4; ++mt)
#pragma unroll
        for (int r = 0; r < 8; ++r)
            O[(size_t)(row0 + mt * 16 + mbase + r) * HH + n] = acc[mt][r];
}

// ---------------------------------------------------------------------------
// Kernel 2: sequential recurrence, batch-parallel (16 blocks x 16 rows).
// Recurrent weights pre-swizzled into WMMA fragment layout in LDS:
//   Wfrag[((g*8+nt)*32+kq)*64 + lane*2 .. +1]  (one conflict-free b64/WMMA).
// State tiles padded to stride 132 floats -> conflict-free A-fragment reads
// and conflict-free per-row cumprod scanners.
// ---------------------------------------------------------------------------
__global__ __launch_bounds__(256, 1)
void qlstm_recur(const float* __restrict__ Wp,
                 const float* __restrict__ Wu,
                 const float* __restrict__ Wo,
                 const float* __restrict__ p_forget,
                 const float* __restrict__ p_input,
                 const float* __restrict__ XP,
                 const float* __restrict__ XU,
                 const float* __restrict__ XO,
                 float* __restrict__ outs,
                 float* __restrict__ hx_out,
                 float* __restrict__ cx_out)
{
    extern __shared__ float smem[];
    float* Wfrag = smem;                   // 3*H*H floats, fragment layout
    float* hbuf  = Wfrag + 3 * HH * HH;    // [16][HP]
    float* cbuf  = hbuf + 16 * HP;         // [16][HP]
    float* qv    = cbuf + 16 * HP;         // [16][HP]
    float* gu    = qv + 16 * HP;           // [16][HP]
    float* go    = gu + 16 * HP;           // [16][HP]
    float* CF    = go + 16 * HP;           // [NQ]
    float* CI    = CF + NQ;                // [NQ]

    const int tid   = threadIdx.x;
    const int brow0 = blockIdx.x * 16;

    // Stage recurrent weights in per-lane fragment layout:
    // i -> (g, nt, kq, l); value = (W[n][D+k], W[n][D+k+1]) for that lane.
    for (int i = tid; i < 3 * 8 * 32 * 32; i += 256) {
        int l  = i & 31;
        int kq = (i >> 5) & 31;
        int nt = (i >> 10) & 7;
        int g  = i >> 13;
        const float* W = (g == 0) ? Wp : (g == 1) ? Wu : Wo;
        int n = nt * 16 + (l & 15);
        int k = 4 * kq + ((l < 16) ? 0 : 2);
        v2f v;
        v.x = W[(size_t)n * DH + DD + k];
        v.y = W[(size_t)n * DH + DD + k + 1];
        *(v2f*)&Wfrag[(size_t)i * 2] = v;
    }
    for (int i = tid; i < 16 * HP; i += 256) { hbuf[i] = 0.0f; cbuf[i] = 0.0f; }
    if (tid == 0)  { float p = 1.f; for (int k = 0; k < NQ; ++k) { p *= __cosf(p_forget[k]); CF[k] = p; } }
    if (tid == 32) { float p = 1.f; for (int k = 0; k < NQ; ++k) { p *= __cosf(p_input[k]);  CI[k] = p; } }
    __syncthreads();

    const int wave  = tid >> 5;            // n-tile 0..7
    const int lane  = tid & 31;
    const int n     = wave * 16 + (lane & 15);
    const int koff  = (lane < 16) ? 0 : 2;
    const int m     = lane & 15;
    const int mbase = (lane < 16) ? 0 : 8;

    for (int t = 0; t < TT; ++t) {
        const size_t rbase = ((size_t)t * BB + brow0) * HH;

        // Stage this step's x-projections (coalesced global, padded LDS).
        for (int i = tid; i < 16 * HH; i += 256) {
            int r = i >> 7, k = i & 127;
            int idx = r * HP + k;
            qv[idx] = XP[rbase + i];
            gu[idx] = XU[rbase + i];
            go[idx] = XO[rbase + i];
        }
        // Prefetch next timestep's rows (global_prefetch_b8).
        if (t + 1 < TT && tid < 192) {
            int arr = tid >> 6;
            size_t off = rbase + (size_t)BB * HH + (size_t)(tid & 63) * 32;
            const float* src = (arr == 0) ? XP : (arr == 1) ? XU : XO;
            __builtin_prefetch(&src[off], 0, 1);
        }
        __syncthreads();

        // Recurrent GEMM: preact += h @ Wh^T (C seeded from staged x-part).
#pragma unroll
        for (int g = 0; g < 3; ++g) {
            float* Bv = (g == 0) ? qv : (g == 1) ? gu : go;
            v8f acc;
#pragma unroll
            for (int r = 0; r < 8; ++r) acc[r] = Bv[(mbase + r) * HP + n];

            const float* WF = &Wfrag[(size_t)((g * 8 + wave) * 32) * 64];
            for (int kq = 0; kq < 32; ++kq) {
                v2f a = *(const v2f*)&hbuf[m * HP + 4 * kq + koff];
                v2f b = *(const v2f*)&WF[kq * 64 + lane * 2];
                acc = __builtin_amdgcn_wmma_f32_16x16x4_f32(
                    false, a, false, b, (short)0, acc, false, false);
            }
#pragma unroll
            for (int r = 0; r < 8; ++r) Bv[(mbase + r) * HP + n] = acc[r];
        }
        __syncthreads();

        // cumprod(cos(qvec)) per row; padded stride -> scanners hit
        // distinct banks.
        if (tid < 16) {
            float p = 1.0f;
            float* row = &qv[tid * HP];
            for (int k = 0; k < NQ; ++k) { p *= __cosf(row[k]); row[k] = p; }
        }
        __syncthreads();

        // Gates + state update; coalesced h_t store.
#pragma unroll
        for (int j = 0; j < 8; ++j) {
            int e   = j * 256 + tid;          // 0..2047
            int r   = e >> 7;
            int k   = e & 127;
            int idx = r * HP + k;
            float cum = qv[idx];
            float f   = sigmoidf_(cum * CF[k]);
            float ig  = sigmoidf_(cum * CI[k]);
            float gg  = tanhf(gu[idx]);
            float oo  = sigmoidf_(go[idx]);
            float cn  = f * cbuf[idx] + ig * gg;
            float hn  = oo * tanhf(cn);
            cbuf[idx] = cn;
            hbuf[idx] = hn;
            outs[(size_t)t * BB * HH + (size_t)brow0 * HH + e] = hn;
        }
        __syncthreads();
    }

    for (int i = tid; i < 16 * HH; i += 256) {
        int idx = (i >> 7) * HP + (i & 127);
        hx_out[(size_t)brow0 * HH + i] = hbuf[idx];
        cx_out[(size_t)brow0 * HH + i] = cbuf[idx];
    }
}

// ---------------------------------------------------------------------------
extern "C" void kernel_launch(void* const* d_in, const int* in_sizes, int n_in,
                              void* d_out, int out_size, void* d_ws, size_t ws_size,
                              hipStream_t stream)
{
    const float* X  = (const float*)d_in[0]; // [T,B,D]
    const float* Wp = (const float*)d_in[1]; // [NQ, D+H]
    const float* pf = (const float*)d_in[2]; // [NQ]
    const float* pi = (const float*)d_in[3]; // [NQ]
    const float* Wu = (const float*)d_in[4]; // [H, D+H]
    const float* bu = (const float*)d_in[5]; // [H]
    const float* Wo = (const float*)d_in[6]; // [H, D+H]
    const float* bo = (const float*)d_in[7]; // [H]

    float* out  = (float*)d_out;
    float* outs = out;                               // [T,B,H]
    float* hxo  = out + (size_t)TT * BB * HH;        // [B,H]
    float* cxo  = hxo + (size_t)BB * HH;             // [B,H]

    float* XP = (float*)d_ws;                        // [T*B, H]
    float* XU = XP + (size_t)TT * BB * HH;
    float* XO = XU + (size_t)TT * BB * HH;

    // Phase 1: parallel x-projections, 2048 blocks x 768 threads,
    // dynamic LDS: 64 * 260 * 4 = 66,560 B.
    const size_t smem1 = (size_t)64 * XS_LD * sizeof(float);
    qlstm_xproj<<<(TT * BB) / 64, 768, smem1, stream>>>(X, Wp, Wu, Wo, bu, bo,
                                                        XP, XU, XO);

    // Phase 2: batch-parallel recurrence, 16 blocks,
    // dynamic LDS: (3*128*128 + 5*16*132 + 2*128) * 4 = 239,872 B (< 320 KB).
    const size_t smem2 =
        (size_t)(3 * HH * HH + 5 * 16 * HP + 2 * NQ) * sizeof(float);
    qlstm_recur<<<BB / 16, 256, smem2, stream>>>(Wp, Wu, Wo, pf, pi,
                                                 XP, XU, XO,
                                                 outs, hxo, cxo);
}